// SetToGraph_17351667876287
// MI455X (gfx1250) — compile-verified
//
#include <hip/hip_runtime.h>

typedef __attribute__((ext_vector_type(2))) float v2f;
typedef __attribute__((ext_vector_type(8))) float v8f;
typedef __attribute__((ext_vector_type(4))) unsigned int u32x4;
typedef __attribute__((ext_vector_type(4))) int i32x4;
typedef __attribute__((ext_vector_type(8))) int i32x8;

// ---------------------------------------------------------------------------
// TDM: async 2D tile load global -> LDS (ISA ch.7/8, "Tensor Data Mover").
// Loads tile_h rows x tile_w f32 elements (row stride = stride_elems) into
// LDS at lds_off, inserting 1 pad DWORD after every 32 DWORDs so the LDS
// image matches a float[rows][33] padded array (bank-conflict-free reads).
// Descriptor packing per CDNA5 ISA 8.3/8.4. Uniform args (SGPR-sourced);
// issue under a wave-uniform branch. Completion via TENSORcnt.
// ---------------------------------------------------------------------------
__device__ __forceinline__ void tdm_load_tile_f32(
    const float* gsrc, unsigned lds_off,
    unsigned tile_w, unsigned tile_h,
    unsigned stride_elems, unsigned tensor_w, unsigned tensor_h) {
  const unsigned long long ga = (unsigned long long)(uintptr_t)gsrc;
  u32x4 g0;
  g0[0] = 1u;                                        // count=1 (valid), user mode
  g0[1] = lds_off;                                   // lds_addr (bytes)
  g0[2] = (unsigned)(ga & 0xFFFFFFFFull);            // global_addr[31:0]
  g0[3] = (unsigned)((ga >> 32) & 0x01FFFFFFull)     // global_addr[56:32]
          | (2u << 30);                              // type=2 ("image")
  i32x8 g1;
  g1[0] = (int)((2u << 16)                           // data_size = 4 bytes
                | (1u << 20)                         // pad_enable
                | (4u << 22)                         // pad_interval: 32 DWORDs
                | (0u << 25));                       // pad_amount: 1 DWORD
  g1[1] = (int)((tensor_w & 0xFFFFu) << 16);         // tensor_dim0[15:0]
  g1[2] = (int)(((tensor_w >> 16) & 0xFFFFu)         // tensor_dim0[31:16]
                | ((tensor_h & 0xFFFFu) << 16));     // tensor_dim1[15:0]
  g1[3] = (int)(((tensor_h >> 16) & 0xFFFFu)         // tensor_dim1[31:16]
                | ((tile_w & 0xFFFFu) << 16));       // tile_dim0
  g1[4] = (int)(tile_h & 0xFFFFu);                   // tile_dim1 (tile_dim2=0)
  g1[5] = (int)stride_elems;                         // tensor_dim0_stride[31:0]
  g1[6] = 0;                                         // stride hi / dim1_stride lo
  g1[7] = 0;
  const i32x4 zero4 = {0, 0, 0, 0};
#if defined(__clang_major__) && (__clang_major__ >= 23)
  const i32x8 zero8 = {0, 0, 0, 0, 0, 0, 0, 0};
  __builtin_amdgcn_tensor_load_to_lds(g0, g1, zero4, zero4, zero8, 0);
#else
  __builtin_amdgcn_tensor_load_to_lds(g0, g1, zero4, zero4, 0);
#endif
}

// ---------------------------------------------------------------------------
// mean over the N axis: one block per (c, b)
// ---------------------------------------------------------------------------
template <int NPOS>
__global__ __launch_bounds__(256) void mean_over_n_kernel(
    const float* __restrict__ in, float* __restrict__ mean,
    int C, long long sB, long long sC, long long sN) {
  const int c = blockIdx.x;
  const int b = blockIdx.y;
  const int tid = threadIdx.x;
  float s = 0.f;
  const long long base = (long long)b * sB + (long long)c * sC;
#pragma unroll
  for (int n = tid; n < NPOS; n += 256) s += in[base + (long long)n * sN];
  __shared__ float red[256];
  red[tid] = s;
  __syncthreads();
  for (int off = 128; off > 0; off >>= 1) {
    if (tid < off) red[tid] += red[tid + off];
    __syncthreads();
  }
  if (tid == 0) mean[b * C + c] = red[0] * (1.f / (float)NPOS);
}

// ---------------------------------------------------------------------------
// DeepSet layer GEMM on V_WMMA_F32_16X16X4_F32, fully compile-time shaped.
//   Hout[b,o,n] = act( sum_k (W1[o,k] (+W2[o,k])) * Hin[b,k,n] + beta[o] )
//   beta[o] = bias1[o] + bias2[o] - sum_k W2[o,k]*mean[b,k]
// One wave per 16x16 (o,n) tile; K unrolled in steps of 4.
// fp32 16x16x4 fragment layouts (ISA 7.12.2):
//   A: lane L -> row m=L&15, K = 2*(L>>4)+{0,1}   (contiguous pair -> v2f load)
//   B: lane L -> col n=L&15, K = 2*(L>>4)+{0,1}
//   C/D: lane L -> col n=L&15, VGPR v -> row m = v + 8*(L>>4)
// XIN: input is x with layout [b][n][CI] (K contiguous); else [b][CI][NPOS].
// ---------------------------------------------------------------------------
template <int CI, int CO, int NPOS, bool HAS_W2, bool RELU, bool XIN>
__global__ __launch_bounds__(32) void deepset_gemm_wmma(
    const float* __restrict__ Hin,
    const float* __restrict__ W1,
    const float* __restrict__ W2,      // used iff HAS_W2
    const float* __restrict__ bias1,   // nullable
    const float* __restrict__ bias2,   // nullable
    const float* __restrict__ mean,    // (B, CI), used iff HAS_W2
    float* __restrict__ Hout,          // (B, CO, NPOS)
    int ldw, int coloff) {
  const int lane = threadIdx.x;        // one wave32 per block
  const int b    = blockIdx.z;
  const int o0   = blockIdx.y * 16;
  const int n0   = blockIdx.x * 16;

  __shared__ float beta[16];
  if (lane < 16) {
    const int o = o0 + lane;
    float v = 0.f;
    if (bias1) v += bias1[o];
    if (bias2) v += bias2[o];
    if (HAS_W2) {
      const float* w2row = W2 + (long long)o * ldw + coloff;
      const float* mrow  = mean + b * CI;
      float acc = 0.f;
#pragma unroll
      for (int c = 0; c < CI; ++c) acc += w2row[c] * mrow[c];
      v -= acc;
    }
    beta[lane] = v;
  }
  __syncthreads();

  const int mrow = lane & 15;          // A row / B column / D column
  const int half = lane >> 4;          // 0 or 1
  const int koff = half * 2;           // this lane's K pair base offset

  const float* a1row = W1 + (long long)(o0 + mrow) * ldw + coloff;
  const float* a2row = HAS_W2 ? (W2 + (long long)(o0 + mrow) * ldw + coloff)
                              : nullptr;
  const float* binX = Hin + ((long long)b * NPOS + (n0 + mrow)) * CI; // + k
  const float* binH = Hin + (long long)b * CI * NPOS + (n0 + mrow);   // + k*NPOS

  v8f acc = {};
#pragma unroll
  for (int k = 0; k < CI; k += 4) {
    const int ka = k + koff;           // ka, ka+1 are this lane's K pair
    v2f afrag, bfrag;
    if (k + 4 <= CI) {                 // statically-known full step
      afrag = *(const v2f*)(a1row + ka);            // 8B-aligned (ka even)
      if (HAS_W2) {
        v2f a2 = *(const v2f*)(a2row + ka);
        afrag.x += a2.x; afrag.y += a2.y;
      }
      if (XIN) {
        bfrag = *(const v2f*)(binX + ka);
      } else {
        bfrag.x = binH[(long long)ka * NPOS];
        bfrag.y = binH[(long long)(ka + 1) * NPOS];
      }
    } else {                           // statically-known remainder (CI%4!=0)
      const int k0 = (ka < CI) ? ka : 0;
      const int k1 = (ka + 1 < CI) ? ka + 1 : 0;
      float a0 = a1row[k0], a1 = a1row[k1];
      if (HAS_W2) { a0 += a2row[k0]; a1 += a2row[k1]; }
      float b0, b1;
      if (XIN) { b0 = binX[k0]; b1 = binX[k1]; }
      else     { b0 = binH[(long long)k0 * NPOS];
                 b1 = binH[(long long)k1 * NPOS]; }
      afrag.x = (ka     < CI) ? a0 : 0.f;
      afrag.y = (ka + 1 < CI) ? a1 : 0.f;
      bfrag.x = (ka     < CI) ? b0 : 0.f;
      bfrag.y = (ka + 1 < CI) ? b1 : 0.f;
    }
    acc = __builtin_amdgcn_wmma_f32_16x16x4_f32(
        false, afrag, false, bfrag, (short)0, acc, false, false);
  }

#pragma unroll
  for (int v = 0; v < 8; ++v) {
    const int mo = v + 8 * half;
    float r = acc[v] + beta[mo];
    if (RELU) r = fmaxf(r, 0.f);
    Hout[((long long)b * CO + (o0 + mo)) * NPOS + (n0 + mrow)] = r;
  }
}

// ---------------------------------------------------------------------------
// Epilogue: out[b,i,j] = sum_c sw1[c]*relu(q[b,c,i] + pb[b,c,j]) + sb1
// (sb0 folded into pb). ReLU sits inside the contraction -> inherently VALU.
// p/q 64x32 tiles staged by the Tensor Data Mover directly into LDS with
// hardware row padding (33-float rows). 2x2 register tile per thread.
// ---------------------------------------------------------------------------
template <int NPOS>
__global__ __launch_bounds__(256) void pairwise_epilogue_kernel(
    const float* __restrict__ pb,   // (B,64,NPOS), includes sb0
    const float* __restrict__ q,    // (B,64,NPOS)
    const float* __restrict__ sw1,  // (64)
    const float* __restrict__ sb1,  // (1)
    float* __restrict__ out) {      // (B,NPOS,NPOS)
  const int H = 64;
  const int b  = blockIdx.z;
  const int i0 = blockIdx.y * 32;
  const int j0 = blockIdx.x * 32;
  const int tid = threadIdx.x;      // 256 threads = 8 waves

  __shared__ float Pt[64][33];      // TDM pads 1 DWORD per 32 -> row stride 33
  __shared__ float Qt[64][33];
  __shared__ float Sw[64];

  if (tid < 64) Sw[tid] = sw1[tid];
  if (tid < 32) {                   // wave 0 issues both DMA descriptors
    tdm_load_tile_f32(pb + (long long)b * H * NPOS + j0,
                      (unsigned)(uintptr_t)&Pt[0][0],
                      /*tile_w=*/32, /*tile_h=*/H,
                      /*stride=*/NPOS, /*tensor_w=*/NPOS, /*tensor_h=*/H);
    tdm_load_tile_f32(q + (long long)b * H * NPOS + i0,
                      (unsigned)(uintptr_t)&Qt[0][0],
                      /*tile_w=*/32, /*tile_h=*/H,
                      /*stride=*/NPOS, /*tensor_w=*/NPOS, /*tensor_h=*/H);
    __builtin_amdgcn_s_wait_tensorcnt(0);
  }
  __syncthreads();

  const int tj = (tid & 15) * 2;
  const int ti = (tid >> 4) * 2;
  float a00 = 0.f, a01 = 0.f, a10 = 0.f, a11 = 0.f;
#pragma unroll 8
  for (int c = 0; c < 64; ++c) {
    const float s  = Sw[c];
    const float pa = Pt[c][tj], pc = Pt[c][tj + 1];
    const float qa = Qt[c][ti], qc = Qt[c][ti + 1];
    a00 += s * fmaxf(qa + pa, 0.f);
    a01 += s * fmaxf(qa + pc, 0.f);
    a10 += s * fmaxf(qc + pa, 0.f);
    a11 += s * fmaxf(qc + pc, 0.f);
  }
  const float bb = sb1[0];
  const long long base = ((long long)b * NPOS + (i0 + ti)) * NPOS + (j0 + tj);
  out[base]            = a00 + bb;
  out[base + 1]        = a01 + bb;
  out[base + NPOS]     = a10 + bb;
  out[base + NPOS + 1] = a11 + bb;
}

// ---------------------------------------------------------------------------
extern "C" void kernel_launch(void* const* d_in, const int* in_sizes, int n_in,
                              void* d_out, int out_size, void* d_ws, size_t ws_size,
                              hipStream_t stream) {
  (void)in_sizes; (void)n_in; (void)out_size; (void)ws_size;
  constexpr int B = 2, N = 768, F_IN = 10;

  const float* x    = (const float*)d_in[0];
  const float* w1_0 = (const float*)d_in[1];
  const float* b1_0 = (const float*)d_in[2];
  const float* w2_0 = (const float*)d_in[3];
  const float* b2_0 = (const float*)d_in[4];
  const float* w1_1 = (const float*)d_in[5];
  const float* b1_1 = (const float*)d_in[6];
  const float* w2_1 = (const float*)d_in[7];
  const float* b2_1 = (const float*)d_in[8];
  const float* w1_2 = (const float*)d_in[9];
  const float* b1_2 = (const float*)d_in[10];
  const float* w2_2 = (const float*)d_in[11];
  const float* b2_2 = (const float*)d_in[12];
  const float* sw0  = (const float*)d_in[13];  // (64,128)
  const float* sb0  = (const float*)d_in[14];  // (64)
  const float* sw1  = (const float*)d_in[15];  // (1,64)
  const float* sb1  = (const float*)d_in[16];  // (1)
  float* out = (float*)d_out;

  float* ws = (float*)d_ws;
  float* h1 = ws;                       // 2*128*768
  float* h2 = h1 + 2 * 128 * 768;       // 2*128*768
  float* u  = h2 + 2 * 128 * 768;       // 2*64*768
  float* p  = u  + 2 * 64 * 768;        // 2*64*768
  float* q  = p  + 2 * 64 * 768;        // 2*64*768
  float* mx = q  + 2 * 64 * 768;        // 2*10 (+pad)
  float* m1 = mx + 32;                  // 2*128
  float* m2 = m1 + 256;                 // 2*128

  // x strides as h[b,c,n] = x[b,n,c]
  const long long xB = (long long)N * F_IN, xC = 1, xN = F_IN;
  const long long h128B = 128LL * N;

  // layer 0: mean(x), then h1 = relu((w1+w2)@h + beta)   [XIN layout]
  mean_over_n_kernel<N><<<dim3(F_IN, B), 256, 0, stream>>>(x, mx, F_IN, xB, xC, xN);
  deepset_gemm_wmma<F_IN, 128, N, true, true, true>
      <<<dim3(N / 16, 128 / 16, B), 32, 0, stream>>>(
          x, w1_0, w2_0, b1_0, b2_0, mx, h1, F_IN, 0);

  // layer 1
  mean_over_n_kernel<N><<<dim3(128, B), 256, 0, stream>>>(h1, m1, 128, h128B, N, 1);
  deepset_gemm_wmma<128, 128, N, true, true, false>
      <<<dim3(N / 16, 128 / 16, B), 32, 0, stream>>>(
          h1, w1_1, w2_1, b1_1, b2_1, m1, h2, 128, 0);

  // layer 2 (no relu)
  mean_over_n_kernel<N><<<dim3(128, B), 256, 0, stream>>>(h2, m2, 128, h128B, N, 1);
  deepset_gemm_wmma<128, 64, N, true, false, false>
      <<<dim3(N / 16, 64 / 16, B), 32, 0, stream>>>(
          h2, w1_2, w2_2, b1_2, b2_2, m2, u, 128, 0);

  // p = sw0[:, :64] @ u + sb0 ; q = sw0[:, 64:] @ u   (ldw=128, column slices)
  deepset_gemm_wmma<64, 64, N, false, false, false>
      <<<dim3(N / 16, 64 / 16, B), 32, 0, stream>>>(
          u, sw0, nullptr, sb0, nullptr, nullptr, p, 128, 0);
  deepset_gemm_wmma<64, 64, N, false, false, false>
      <<<dim3(N / 16, 64 / 16, B), 32, 0, stream>>>(
          u, sw0, nullptr, nullptr, nullptr, nullptr, q, 128, 64);

  // out[b,i,j] = sum_c sw1[c]*relu(q[c,i]+p[c,j]) + sb1
  pairwise_epilogue_kernel<N><<<dim3(N / 32, N / 32, B), 256, 0, stream>>>(
      p, q, sw1, sb1, out);
}